// Net_75282186764765
// MI455X (gfx1250) — compile-verified
//
#include <hip/hip_runtime.h>
#include <math.h>

#define DIM 64
#define OUTC 256

typedef __attribute__((ext_vector_type(2))) float v2f;
typedef __attribute__((ext_vector_type(8))) float v8f;

// ---------------------------------------------------------------------------
// dst[n,:] = src[n,:] / max(||src[n,:]||, 1e-12)   (one 64-thread block / row)
__global__ void row_l2norm_kernel(const float* __restrict__ src,
                                  float* __restrict__ dst) {
  int n = blockIdx.x, t = threadIdx.x;
  size_t base = (size_t)n * DIM;
  float v = src[base + t];
  float ss = v * v;
#pragma unroll
  for (int off = 16; off > 0; off >>= 1) ss += __shfl_down(ss, off, 32);
  __shared__ float red[2];
  if ((t & 31) == 0) red[t >> 5] = ss;
  __syncthreads();
  float r = 1.0f / fmaxf(sqrtf(red[0] + red[1]), 1e-12f);
  dst[base + t] = v * r;
}

// ---------------------------------------------------------------------------
// x[node0+row,:] = l2norm(leaky_relu(feat[row,:] @ W^T + b, 0.01))
__global__ void mlp_feat_kernel(const float* __restrict__ feat,
                                const float* __restrict__ W,
                                const float* __restrict__ bias, int K,
                                int node0, float* __restrict__ xout) {
  __shared__ float srow[128];
  __shared__ float red[2];
  int row = blockIdx.x, t = threadIdx.x;
  for (int k = t; k < K; k += DIM) srow[k] = feat[(size_t)row * K + k];
  __syncthreads();
  float acc = bias[t];
  const float* wr = W + (size_t)t * K;
  for (int k = 0; k < K; ++k) acc = fmaf(srow[k], wr[k], acc);
  acc = acc > 0.f ? acc : 0.01f * acc;
  float ss = acc * acc;
#pragma unroll
  for (int off = 16; off > 0; off >>= 1) ss += __shfl_down(ss, off, 32);
  if ((t & 31) == 0) red[t >> 5] = ss;
  __syncthreads();
  float r = 1.0f / fmaxf(sqrtf(red[0] + red[1]), 1e-12f);
  xout[(size_t)(node0 + row) * DIM + t] = acc * r;
}

// ---------------------------------------------------------------------------
// Y[M,64] = X[M,64] @ W[64,64], full f32 via V_WMMA_F32_16X16X4_F32.
// One wave per 16x16 output tile; K-loop of 16 WMMAs.
// A frag (16x4 f32): lanes 0-15 -> M rows, half 0 holds K{k0,k0+1}, half 1
// K{k0+2,k0+3}. B frag (4x16): VGPR r holds row K=r (+2 for upper lane half).
// C/D: VGPR r -> M=r (lanes 0-15) / M=r+8 (lanes 16-31), N=lane&15.
__global__ void gemm_n64_wmma(const float* __restrict__ X,
                              const float* __restrict__ W,
                              float* __restrict__ Y, int M) {
  int wave = threadIdx.x >> 5;
  int lane = threadIdx.x & 31;
  int tile = blockIdx.x * 8 + wave;
  int mt = tile >> 2, nt = tile & 3;
  if (mt * 16 >= M) return;  // wave-uniform; M is a multiple of 16 here
  int half = lane >> 4;
  int l15 = lane & 15;
  const float* xrow = X + (size_t)(mt * 16 + l15) * 64;
  int ncol = nt * 16 + l15;
  v8f c = {};
#pragma unroll
  for (int k0 = 0; k0 < 64; k0 += 4) {
    int ka = k0 + half * 2;
    v2f a, b;
    a.x = xrow[ka];
    a.y = xrow[ka + 1];
    b.x = W[ka * 64 + ncol];
    b.y = W[(ka + 1) * 64 + ncol];
    c = __builtin_amdgcn_wmma_f32_16x16x4_f32(false, a, false, b, (short)0, c,
                                              false, false);
  }
  float* yb = Y + (size_t)(mt * 16 + half * 8) * 64 + ncol;
#pragma unroll
  for (int r = 0; r < 8; ++r) yb[(size_t)r * 64] = c[r];
}

// ---------------------------------------------------------------------------
__device__ __forceinline__ void edge_sd(const int* __restrict__ edge, int e,
                                        int E, int& s, int& d) {
  s = edge[e];
  d = (e < E) ? edge[e + E] : edge[e - E];
}

// d[e] = dot64(h[dst], h[src])  (raw, for cross-modal gates)
__global__ void edge_dot_kernel(const int* __restrict__ edge, int E, int E2,
                                const float* __restrict__ h,
                                float* __restrict__ dout) {
  int e = blockIdx.x * blockDim.x + threadIdx.x;
  if (e >= E2) return;
  int s, d;
  edge_sd(edge, e, E, s, d);
  const float4* a = (const float4*)(h + (size_t)s * DIM);
  const float4* b = (const float4*)(h + (size_t)d * DIM);
  float acc = 0.f;
#pragma unroll
  for (int i = 0; i < 16; ++i) {
    float4 x = a[i], y = b[i];
    acc += x.x * y.x + x.y * y.y + x.z * y.z + x.w * y.w;
  }
  dout[e] = acc;
}

// ee[e] = exp(leaky(dot64(h[dst],h[src]), 0.2));  ssum[dst] += ee[e]
__global__ void edge_expsum_kernel(const int* __restrict__ edge, int E, int E2,
                                   const float* __restrict__ h,
                                   float* __restrict__ ee,
                                   float* __restrict__ ssum) {
  int e = blockIdx.x * blockDim.x + threadIdx.x;
  if (e >= E2) return;
  int s, d;
  edge_sd(edge, e, E, s, d);
  const float4* a = (const float4*)(h + (size_t)s * DIM);
  const float4* b = (const float4*)(h + (size_t)d * DIM);
  float acc = 0.f;
#pragma unroll
  for (int i = 0; i < 16; ++i) {
    float4 x = a[i], y = b[i];
    acc += x.x * y.x + x.y * y.y + x.z * y.z + x.w * y.w;
  }
  acc = acc > 0.f ? acc : 0.2f * acc;  // GAT attention leaky slope
  float val = expf(acc);
  ee[e] = val;
  atomicAdd(&ssum[d], val);
}

// agg[dst,:] += alpha * h[src,:]; 16 lanes per edge, float4 gather + 4 atomics.
// ee==nullptr -> alpha = 1 (plain GCN conv). g1!=nullptr -> cross-modal gate.
__global__ void edge_scatter_kernel(const int* __restrict__ edge, int E, int E2,
                                    const float* __restrict__ h,
                                    const float* __restrict__ ee,
                                    const float* __restrict__ ssum,
                                    const float* __restrict__ g1,
                                    const float* __restrict__ g2,
                                    float* __restrict__ agg) {
  int t = blockIdx.x * blockDim.x + threadIdx.x;
  int e = t >> 4;
  if (e >= E2) return;
  int sub = t & 15;
  int s, d;
  edge_sd(edge, e, E, s, d);
  float alpha = 1.f;
  if (ee) {
    alpha = ee[e] / (ssum[d] + 1e-16f);
    if (g1) {
      float gate = 1.f / (1.f + expf(-0.5f * (g1[e] + g2[e])));
      if (gate > 0.5f) alpha *= gate;  // ALPHA_THR
    }
  }
  float4 hv = ((const float4*)(h + (size_t)s * DIM))[sub];
  float* dp = agg + (size_t)d * DIM + sub * 4;
  atomicAdd(dp + 0, alpha * hv.x);
  atomicAdd(dp + 1, alpha * hv.y);
  atomicAdd(dp + 2, alpha * hv.z);
  atomicAdd(dp + 3, alpha * hv.w);
}

// x = l2norm(agg) (addx=0) or l2norm(x + agg) (addx=1)
__global__ void add_norm_kernel(float* __restrict__ x,
                                const float* __restrict__ agg, int addx) {
  int n = blockIdx.x, t = threadIdx.x;
  size_t base = (size_t)n * DIM;
  float v = agg[base + t] + (addx ? x[base + t] : 0.f);
  float ss = v * v;
#pragma unroll
  for (int off = 16; off > 0; off >>= 1) ss += __shfl_down(ss, off, 32);
  __shared__ float red[2];
  if ((t & 31) == 0) red[t >> 5] = ss;
  __syncthreads();
  float r = 1.0f / fmaxf(sqrtf(red[0] + red[1]), 1e-12f);
  x[base + t] = v * r;
}

// out[n, col0:col0+64] = x[n,:]
__global__ void write_out_kernel(const float* __restrict__ x,
                                 float* __restrict__ out, int col0, int N) {
  int i = blockIdx.x * blockDim.x + threadIdx.x;
  if (i >= N * DIM) return;
  int n = i >> 6, d = i & 63;
  out[(size_t)n * OUTC + col0 + d] = x[i];
}

// v = leaky(agg, slope); optionally h1 = v; out[n, 0:64] += v
__global__ void gcn_accum_kernel(const float* __restrict__ agg,
                                 float* __restrict__ out,
                                 float* __restrict__ h1, float slope, int N) {
  int i = blockIdx.x * blockDim.x + threadIdx.x;
  if (i >= N * DIM) return;
  float v = agg[i];
  v = v > 0.f ? v : slope * v;
  if (h1) h1[i] = v;
  int n = i >> 6, d = i & 63;
  out[(size_t)n * OUTC + d] += v;
}

__global__ void zero_kernel(float* __restrict__ p, size_t n) {
  size_t i = (size_t)blockIdx.x * blockDim.x + threadIdx.x;
  size_t stride = (size_t)gridDim.x * blockDim.x;
  for (; i < n; i += stride) p[i] = 0.f;
}

// ---------------------------------------------------------------------------
extern "C" void kernel_launch(void* const* d_in, const int* in_sizes, int n_in,
                              void* d_out, int out_size, void* d_ws,
                              size_t ws_size, hipStream_t stream) {
  const int* edge = (const int*)d_in[0];
  const float* v_feat = (const float*)d_in[1];
  const float* a_feat = (const float*)d_in[2];
  const float* t_feat = (const float*)d_in[3];
  const float* pref_v = (const float*)d_in[4];
  const float* pref_a = (const float*)d_in[5];
  const float* pref_t = (const float*)d_in[6];
  const float* mlp_v_w = (const float*)d_in[7];
  const float* mlp_v_b = (const float*)d_in[8];
  const float* mlp_a_w = (const float*)d_in[9];
  const float* mlp_a_b = (const float*)d_in[10];
  const float* mlp_t_w = (const float*)d_in[11];
  const float* mlp_t_b = (const float*)d_in[12];
  const float* pre_w = (const float*)d_in[13];
  const float* ori_w = (const float*)d_in[14];
  const float* gcn_w = (const float*)d_in[15];
  const float* id_emb = (const float*)d_in[16];

  const int E = in_sizes[0] / 2;
  const int E2 = 2 * E;
  const int Kv = in_sizes[7] / DIM;   // 128
  const int Ka = in_sizes[9] / DIM;   // 128
  const int Kt = in_sizes[11] / DIM;  // 100
  const int nUser = in_sizes[4] / DIM;
  const int nItem = in_sizes[1] / Kv;
  const int N = nUser + nItem;

  float* ws = (float*)d_ws;
  const size_t NODEF = (size_t)N * DIM;
  float* xv = ws;            // modality embeddings, updated in place
  float* xa = xv + NODEF;
  float* xt = xa + NODEF;
  float* h = xt + NODEF;     // x @ W
  float* agg = h + NODEF;    // scatter accumulator
  float* dv = agg + NODEF;   // per-edge modality dots
  float* da = dv + E2;
  float* dt = da + E2;
  float* ee = dt + E2;       // per-edge exp(score)
  float* ssum = ee + E2;     // per-dst softmax denominator
  float* out = (float*)d_out;

  const int edgeGrid = (E2 + 255) / 256;
  const int scatGrid = (int)(((long long)E2 * 16 + 255) / 256);
  const int gemmGrid = (((N + 15) / 16) * 4 + 7) / 8;
  const int nfGrid = (int)((NODEF + 255) / 256);

  // normalized prefs -> user rows; feature MLPs -> item rows
  row_l2norm_kernel<<<nUser, DIM, 0, stream>>>(pref_v, xv);
  row_l2norm_kernel<<<nUser, DIM, 0, stream>>>(pref_a, xa);
  row_l2norm_kernel<<<nUser, DIM, 0, stream>>>(pref_t, xt);
  mlp_feat_kernel<<<nItem, DIM, 0, stream>>>(v_feat, mlp_v_w, mlp_v_b, Kv, nUser, xv);
  mlp_feat_kernel<<<nItem, DIM, 0, stream>>>(a_feat, mlp_a_w, mlp_a_b, Ka, nUser, xa);
  mlp_feat_kernel<<<nItem, DIM, 0, stream>>>(t_feat, mlp_t_w, mlp_t_b, Kt, nUser, xt);

  // per-edge modality dots (consumed by gat_pre gates, use INITIAL embeddings)
  edge_dot_kernel<<<edgeGrid, 256, 0, stream>>>(edge, E, E2, xv, dv);
  edge_dot_kernel<<<edgeGrid, 256, 0, stream>>>(edge, E, E2, xa, da);
  edge_dot_kernel<<<edgeGrid, 256, 0, stream>>>(edge, E, E2, xt, dt);

  auto gat_layer = [&](float* x, const float* Wm, const float* g1,
                       const float* g2, int addx) {
    gemm_n64_wmma<<<gemmGrid, 256, 0, stream>>>(x, Wm, h, N);
    zero_kernel<<<256, 256, 0, stream>>>(ssum, (size_t)N);
    zero_kernel<<<1024, 256, 0, stream>>>(agg, NODEF);
    edge_expsum_kernel<<<edgeGrid, 256, 0, stream>>>(edge, E, E2, h, ee, ssum);
    edge_scatter_kernel<<<scatGrid, 256, 0, stream>>>(edge, E, E2, h, ee, ssum,
                                                      g1, g2, agg);
    add_norm_kernel<<<N, DIM, 0, stream>>>(x, agg, addx);
  };

  // modality v: y=a, z=t
  gat_layer(xv, pre_w, da, dt, 0);
  for (int r = 0; r < 3; ++r) gat_layer(xv, ori_w, nullptr, nullptr, 1);
  write_out_kernel<<<nfGrid, 256, 0, stream>>>(xv, out, 64, N);
  // modality a: y=v, z=t
  gat_layer(xa, pre_w + 4096, dv, dt, 0);
  for (int r = 0; r < 3; ++r) gat_layer(xa, ori_w + 4096, nullptr, nullptr, 1);
  write_out_kernel<<<nfGrid, 256, 0, stream>>>(xa, out, 128, N);
  // modality t: y=a, z=v
  gat_layer(xt, pre_w + 8192, da, dv, 0);
  for (int r = 0; r < 3; ++r) gat_layer(xt, ori_w + 8192, nullptr, nullptr, 1);
  write_out_kernel<<<nfGrid, 256, 0, stream>>>(xt, out, 192, N);

  // real_gcn: id_rep = x0 + h1 + leaky(gcn(h1,W1)) + gcn(h1,W2); reuse xv/xa
  row_l2norm_kernel<<<N, DIM, 0, stream>>>(id_emb, xv);  // x0
  write_out_kernel<<<nfGrid, 256, 0, stream>>>(xv, out, 0, N);
  auto gcn_layer = [&](const float* xin, const float* Wg, float* h1out,
                       float slope) {
    gemm_n64_wmma<<<gemmGrid, 256, 0, stream>>>(xin, Wg, h, N);
    zero_kernel<<<1024, 256, 0, stream>>>(agg, NODEF);
    edge_scatter_kernel<<<scatGrid, 256, 0, stream>>>(
        edge, E, E2, h, nullptr, nullptr, nullptr, nullptr, agg);
    gcn_accum_kernel<<<nfGrid, 256, 0, stream>>>(agg, out, h1out, slope, N);
  };
  gcn_layer(xv, gcn_w, xa, 0.01f);                 // h1 -> xa, out += h1
  gcn_layer(xa, gcn_w + 4096, nullptr, 0.01f);     // out += h2
  gcn_layer(xa, gcn_w + 8192, nullptr, 1.0f);      // out += h3 (no act)
}